// CRW_28922309771496
// MI455X (gfx1250) — compile-verified
//
#include <hip/hip_runtime.h>

typedef float v2f __attribute__((ext_vector_type(2)));
typedef float v8f __attribute__((ext_vector_type(8)));
typedef int   v4i __attribute__((vector_size(4 * sizeof(int))));

namespace {
constexpr int kB  = 16;
constexpr int kN  = 196;
constexpr int kNP = 208;             // 196 padded to 13*16 (16B-aligned rows)
constexpr int kC  = 512;
constexpr int kT  = 20;
constexpr int kHD = 128;
constexpr int kBN = kB * kN;         // 3136
constexpr int kM  = kBN * kT;        // 62720
constexpr int kSteps = 2 * kT - 2;   // 38
constexpr float kInvTemp = 1.0f / 0.07f;
constexpr float kEps = 1e-20f;
constexpr size_t kQElems = (size_t)kB * kHD * kT * kN;  // 8028160
}

// ---------------------------------------------------------------------------
// CDNA5 async global->LDS copy (ASYNCcnt path) with safe fallback.
// Probed signature: (v4i AS1* src, v4i AS3* dst, imm offset, imm cpol).
// ---------------------------------------------------------------------------
#if __has_builtin(__builtin_amdgcn_global_load_async_to_lds_b128)
#define HAVE_ASYNC_LDS 1
#else
#define HAVE_ASYNC_LDS 0
#endif

__device__ __forceinline__ void async_cp16(const float* gsrc, float* lds_dst) {
#if HAVE_ASYNC_LDS
  float* g = const_cast<float*>(gsrc);
  __builtin_amdgcn_global_load_async_to_lds_b128(
      (__attribute__((address_space(1))) v4i*)g,
      (__attribute__((address_space(3))) v4i*)lds_dst, 0, 0);
#else
  *reinterpret_cast<float4*>(lds_dst) = *reinterpret_cast<const float4*>(gsrc);
#endif
}

__device__ __forceinline__ void wait_async_all() {
#if __has_builtin(__builtin_amdgcn_s_wait_asynccnt)
  __builtin_amdgcn_s_wait_asynccnt((short)0);
#else
  asm volatile("s_wait_asynccnt 0" ::: "memory");
#endif
}

// ---------------------------------------------------------------------------
// Kernel 1: fused pool + head GEMM (f32 WMMA) + bias + L2 normalize.
// Block: 256 threads (8 waves). Tile: 64 rows (m = bn*T+t) x 128 cols (d).
// ---------------------------------------------------------------------------
__global__ __launch_bounds__(256) void head_pool_gemm_norm_kernel(
    const float* __restrict__ maps, const float* __restrict__ Wh,
    const float* __restrict__ bh, float* __restrict__ q_out,
    float* __restrict__ qn)
{
  __shared__ float As[64][33];
  __shared__ float Ws[32][130];
  __shared__ float Cs[64][129];
  __shared__ float ps[256];
  __shared__ float rn[64];

  const int tid  = threadIdx.x;
  const int lane = tid & 31;
  const int wv   = tid >> 5;
  const int m0   = blockIdx.x * 64;

  v8f acc[4] = {};
  const int mt  = wv >> 1;        // 0..3 : which 16-row M tile
  const int ntb = (wv & 1) * 4;   // 0 or 4 : first of four 16-col N tiles
  const int row = lane & 15;
  const int kh  = (lane >> 4) * 2;

  for (int k0 = 0; k0 < kC; k0 += 32) {
    // Stage A chunk (64x32), pooling maps' h*w=4 on the fly (float4 loads).
    for (int i = tid; i < 64 * 32; i += 256) {
      const int r  = i >> 5;
      const int kc = i & 31;
      const int m  = m0 + r;
      const int bn = m / kT;
      const int t  = m - bn * kT;
      const float4 v = *reinterpret_cast<const float4*>(
          maps + ((((size_t)bn * kC + (k0 + kc)) * kT + t) << 2));
      As[r][kc] = 0.25f * (v.x + v.y + v.z + v.w);
    }
    // Stage W chunk (32x128).
    for (int i = tid; i < 32 * kHD; i += 256) {
      const int r = i >> 7;
      const int d = i & 127;
      Ws[r][d] = Wh[(size_t)(k0 + r) * kHD + d];
    }
    __syncthreads();
#pragma unroll
    for (int kk = 0; kk < 32; kk += 4) {
      v2f a;
      a.x = As[mt * 16 + row][kk + kh];
      a.y = As[mt * 16 + row][kk + kh + 1];
#pragma unroll
      for (int j = 0; j < 4; ++j) {
        const int n0 = (ntb + j) * 16;
        v2f bf;
        bf.x = Ws[kk + kh][n0 + row];
        bf.y = Ws[kk + kh + 1][n0 + row];
        acc[j] = __builtin_amdgcn_wmma_f32_16x16x4_f32(
            false, a, false, bf, (short)0, acc[j], false, false);
      }
    }
    __syncthreads();
  }

  // Spill accumulators (+bias) to LDS so full rows are visible for the norm.
#pragma unroll
  for (int j = 0; j < 4; ++j) {
    const int col  = (ntb + j) * 16 + row;
    const float bias = bh[col];
#pragma unroll
    for (int r = 0; r < 8; ++r)
      Cs[mt * 16 + r + 8 * (lane >> 4)][col] = acc[j][r] + bias;
  }
  __syncthreads();

  // Row L2 norms: 4 threads per row, each sums 32 squares.
  {
    const int r = tid >> 2, part = tid & 3;
    float s = 0.f;
    for (int d = part * 32; d < part * 32 + 32; ++d) {
      const float v = Cs[r][d];
      s += v * v;
    }
    ps[tid] = s;
    __syncthreads();
    if (part == 0) {
      const float tot = ps[tid] + ps[tid + 1] + ps[tid + 2] + ps[tid + 3];
      rn[r] = 1.0f / fmaxf(sqrtf(tot), 1e-12f);
    }
  }
  __syncthreads();

  // Write normalized q twice: output layout (b,d,t,n) and GEMM layout (b,t,n,d).
  for (int i = tid; i < 64 * kHD; i += 256) {
    const int r = i >> 7, d = i & 127;
    const int m  = m0 + r;
    const int bn = m / kT;
    const int t  = m - bn * kT;
    const int b  = bn / kN;
    const int n  = bn - b * kN;
    const float v = Cs[r][d] * rn[r];
    q_out[(((size_t)b * kHD + d) * kT + t) * kN + n] = v;
    qn[(((size_t)b * kT + t) * kNP + n) * kHD + d]   = v;  // padded token dim
  }
}

// ---------------------------------------------------------------------------
// Kernel 2: A_t = Q_t * Q_{t+1}^T / temp (NT GEMM, K=128), written to slot t
// and transposed into slot 37-t. Async double-buffered LDS staging.
// qn pad rows (196..207) are zero, so AA's pad region ends up zero.
// ---------------------------------------------------------------------------
__global__ __launch_bounds__(256) void affinity_kernel(
    const float* __restrict__ qn, float* __restrict__ AA)
{
  __shared__ float Qa[2][64][20];
  __shared__ float Qb[2][64][20];

  const int tid  = threadIdx.x;
  const int lane = tid & 31;
  const int wv   = tid >> 5;
  const int z = blockIdx.z;
  const int b = z / (kT - 1);
  const int t = z - b * (kT - 1);
  const int i0 = blockIdx.y * 64;
  const int j0 = blockIdx.x * 64;
  const float* Qt  = qn + ((size_t)b * kT + t) * kNP * kHD;
  const float* Qt1 = Qt + (size_t)kNP * kHD;

  v8f acc[2] = {};
  const int row = lane & 15;
  const int kh  = (lane >> 4) * 2;

  auto stage = [&](int c, int pb) {
    const int k0 = c * 16;
    const int r  = tid >> 2;            // 0..63
    const int c4 = (tid & 3) * 4;       // 0,4,8,12
    const int gi = i0 + r;  const int gsi = (gi < kNP) ? gi : (kNP - 1);
    const int gj = j0 + r;  const int gsj = (gj < kNP) ? gj : (kNP - 1);
    async_cp16(Qt  + (size_t)gsi * kHD + k0 + c4, &Qa[pb][r][c4]);
    async_cp16(Qt1 + (size_t)gsj * kHD + k0 + c4, &Qb[pb][r][c4]);
  };
  auto compute = [&](int pb) {
#pragma unroll
    for (int e = 0; e < 2; ++e) {
      const int idx = wv * 2 + e;
      const int mt = idx >> 2, nt = idx & 3;
#pragma unroll
      for (int kk = 0; kk < 16; kk += 4) {
        v2f a, bf;
        a.x  = Qa[pb][mt * 16 + row][kk + kh];
        a.y  = Qa[pb][mt * 16 + row][kk + kh + 1];
        bf.x = Qb[pb][nt * 16 + row][kk + kh];   // B[k][col] = Qt1[col][k]
        bf.y = Qb[pb][nt * 16 + row][kk + kh + 1];
        acc[e] = __builtin_amdgcn_wmma_f32_16x16x4_f32(
            false, a, false, bf, (short)0, acc[e], false, false);
      }
    }
  };

  stage(0, 0);
  for (int c = 0; c < kHD / 16; ++c) {     // 8 chunks
    wait_async_all();
    __syncthreads();
    if (c + 1 < kHD / 16) stage(c + 1, (c + 1) & 1);
    compute(c & 1);
    __syncthreads();
  }

#pragma unroll
  for (int e = 0; e < 2; ++e) {
    const int idx = wv * 2 + e;
    const int mt = idx >> 2, nt = idx & 3;
    const int gj = j0 + nt * 16 + row;
    if (gj < kNP) {
#pragma unroll
      for (int r = 0; r < 8; ++r) {
        const int gi = i0 + mt * 16 + r + 8 * (lane >> 4);
        if (gi < kNP) {
          const float v = acc[e][r] * kInvTemp;
          AA[(((size_t)b * kSteps + t) * kNP + gi) * kNP + gj] = v;
          AA[(((size_t)b * kSteps + (kSteps - 1 - t)) * kNP + gj) * kNP + gi] = v;
        }
      }
    }
  }
}

// ---------------------------------------------------------------------------
// Kernel 3: in-place row softmax over first 196 columns (real rows only).
// Pad rows/cols remain zero.
// ---------------------------------------------------------------------------
__global__ __launch_bounds__(256) void softmax_kernel(float* __restrict__ AA)
{
  __shared__ float red[256];
  const int bt = blockIdx.x / kN;
  const int i  = blockIdx.x - bt * kN;
  float* p = AA + ((size_t)bt * kNP + i) * kNP;
  const int j = threadIdx.x;
  const float v = (j < kN) ? p[j] : -3.4e38f;
  red[j] = v;
  __syncthreads();
  for (int s = 128; s > 0; s >>= 1) {
    if (j < s) red[j] = fmaxf(red[j], red[j + s]);
    __syncthreads();
  }
  const float mx = red[0];
  __syncthreads();
  const float e = (j < kN) ? __expf(v - mx) : 0.f;
  red[j] = e;
  __syncthreads();
  for (int s = 128; s > 0; s >>= 1) {
    if (j < s) red[j] += red[j + s];
    __syncthreads();
  }
  const float inv = 1.0f / red[0];
  if (j < kN) p[j] = e * inv;
}

// ---------------------------------------------------------------------------
// Kernel 4: At0 = identity on the real 196x196 block, zero padding elsewhere.
// ---------------------------------------------------------------------------
__global__ __launch_bounds__(256) void eye_kernel(float* __restrict__ At)
{
  const size_t i = (size_t)blockIdx.x * 256 + threadIdx.x;  // over B*NP*NP
  const int col = (int)(i % kNP);
  const int r   = (int)((i / kNP) % kNP);
  At[i] = (r == col && r < kN) ? 1.f : 0.f;
}

// ---------------------------------------------------------------------------
// Kernel 5: one chain step Anew = P[tstep] @ Aold (NN GEMM, batched over b).
// Async double-buffered LDS staging; K padded to 208 (pad contributes zero).
// ---------------------------------------------------------------------------
__global__ __launch_bounds__(256) void chain_step_kernel(
    const float* __restrict__ AA, int tstep,
    const float* __restrict__ Aold, float* __restrict__ Anew)
{
  __shared__ float Ps[2][64][20];
  __shared__ float Bs[2][16][68];

  const int tid  = threadIdx.x;
  const int lane = tid & 31;
  const int wv   = tid >> 5;
  const int b  = blockIdx.z;
  const int i0 = blockIdx.y * 64;
  const int j0 = blockIdx.x * 64;
  const float* P  = AA + ((size_t)b * kSteps + tstep) * kNP * kNP;
  const float* Ao = Aold + (size_t)b * kNP * kNP;
  float* An = Anew + (size_t)b * kNP * kNP;

  v8f acc[2] = {};
  const int row = lane & 15;
  const int kh  = (lane >> 4) * 2;

  auto stage = [&](int c, int pb) {
    const int k0 = c * 16;
    {  // P tile: 64 rows x 16 k-cols
      const int r  = tid >> 2;
      const int c4 = (tid & 3) * 4;
      const int gi = i0 + r;
      const int gsi = (gi < kNP) ? gi : (kNP - 1);
      async_cp16(P + (size_t)gsi * kNP + k0 + c4, &Ps[pb][r][c4]);
    }
    {  // Aold tile: 16 k-rows x 64 cols
      const int kr = tid >> 4;            // 0..15
      const int c4 = (tid & 15) * 4;      // 0..60
      const int gj = j0 + c4;
      const int gsj = (gj <= kNP - 4) ? gj : (kNP - 4);
      async_cp16(Ao + (size_t)(k0 + kr) * kNP + gsj, &Bs[pb][kr][c4]);
    }
  };
  auto compute = [&](int pb) {
#pragma unroll
    for (int e = 0; e < 2; ++e) {
      const int idx = wv * 2 + e;
      const int mt = idx >> 2, nt = idx & 3;
#pragma unroll
      for (int kk = 0; kk < 16; kk += 4) {
        v2f a, bf;
        a.x  = Ps[pb][mt * 16 + row][kk + kh];
        a.y  = Ps[pb][mt * 16 + row][kk + kh + 1];
        bf.x = Bs[pb][kk + kh][nt * 16 + row];
        bf.y = Bs[pb][kk + kh + 1][nt * 16 + row];
        acc[e] = __builtin_amdgcn_wmma_f32_16x16x4_f32(
            false, a, false, bf, (short)0, acc[e], false, false);
      }
    }
  };

  stage(0, 0);
  for (int c = 0; c < kNP / 16; ++c) {     // 13 chunks
    wait_async_all();
    __syncthreads();
    if (c + 1 < kNP / 16) stage(c + 1, (c + 1) & 1);
    compute(c & 1);
    __syncthreads();
  }

#pragma unroll
  for (int e = 0; e < 2; ++e) {
    const int idx = wv * 2 + e;
    const int mt = idx >> 2, nt = idx & 3;
    const int gj = j0 + nt * 16 + row;
    if (gj < kNP) {
#pragma unroll
      for (int r = 0; r < 8; ++r) {
        const int gi = i0 + mt * 16 + r + 8 * (lane >> 4);
        if (gi < kNP) An[(size_t)gi * kNP + gj] = acc[e][r];
      }
    }
  }
}

// ---------------------------------------------------------------------------
// Kernel 6: per-row cross-entropy term + argmax-correct flag.
// ---------------------------------------------------------------------------
__global__ __launch_bounds__(256) void loss_row_kernel(
    const float* __restrict__ At, float* __restrict__ pl, float* __restrict__ pa)
{
  __shared__ float rv[256];
  __shared__ int   ri[256];
  __shared__ float diag;
  const int rowid = blockIdx.x;         // b*N + i
  const int b = rowid / kN;
  const int i = rowid - b * kN;
  const float* r = At + ((size_t)b * kNP + i) * kNP;
  const int j = threadIdx.x;
  const float v = (j < kN) ? logf(r[j] + kEps) : -3.4e38f;
  if (j == i) diag = v;
  rv[j] = v;
  ri[j] = (j < kN) ? j : (1 << 20);
  __syncthreads();
  for (int s = 128; s > 0; s >>= 1) {
    if (j < s) {
      const float vo = rv[j + s]; const int io = ri[j + s];
      if (vo > rv[j] || (vo == rv[j] && io < ri[j])) { rv[j] = vo; ri[j] = io; }
    }
    __syncthreads();
  }
  const float mx = rv[0];
  const int amax = ri[0];
  __syncthreads();
  rv[j] = (j < kN) ? __expf(v - mx) : 0.f;
  __syncthreads();
  for (int s = 128; s > 0; s >>= 1) {
    if (j < s) rv[j] += rv[j + s];
    __syncthreads();
  }
  if (j == 0) {
    pl[rowid] = (mx + logf(rv[0])) - diag;   // logsumexp - target logit
    pa[rowid] = (amax == i) ? 1.f : 0.f;
  }
}

// ---------------------------------------------------------------------------
// Kernel 7: deterministic final reduction -> loss, acc.
// ---------------------------------------------------------------------------
__global__ __launch_bounds__(256) void loss_reduce_kernel(
    const float* __restrict__ pl, const float* __restrict__ pa,
    float* __restrict__ out)
{
  __shared__ float s1[256], s2[256];
  float a = 0.f, c = 0.f;
  for (int i = threadIdx.x; i < kBN; i += 256) { a += pl[i]; c += pa[i]; }
  s1[threadIdx.x] = a; s2[threadIdx.x] = c;
  __syncthreads();
  for (int s = 128; s > 0; s >>= 1) {
    if (threadIdx.x < s) {
      s1[threadIdx.x] += s1[threadIdx.x + s];
      s2[threadIdx.x] += s2[threadIdx.x + s];
    }
    __syncthreads();
  }
  if (threadIdx.x == 0) {
    out[0] = s1[0] / (float)kBN;
    out[1] = s2[0] / (float)kBN;
  }
}

// ---------------------------------------------------------------------------
extern "C" void kernel_launch(void* const* d_in, const int* in_sizes, int n_in,
                              void* d_out, int out_size, void* d_ws, size_t ws_size,
                              hipStream_t stream) {
  (void)in_sizes; (void)n_in; (void)out_size; (void)ws_size;
  const float* maps = (const float*)d_in[0];
  const float* Wh   = (const float*)d_in[1];
  const float* bh   = (const float*)d_in[2];
  float* out = (float*)d_out;

  // Workspace layout (floats): ~138 MB total.
  const size_t qnElems = (size_t)kB * kT * kNP * kHD;       //  8,519,680
  const size_t aaElems = (size_t)kB * kSteps * kNP * kNP;   // 26,304,512
  const size_t atElems = (size_t)kB * kNP * kNP;            //    692,224
  float* qn  = (float*)d_ws;
  float* AA  = qn  + qnElems;
  float* Ata = AA  + aaElems;
  float* Atb = Ata + atElems;
  float* pl  = Atb + atElems;
  float* pa  = pl  + kBN;

  // Zero qn so its pad rows (n in [196,208)) are zero every call.
  (void)hipMemsetAsync(qn, 0, qnElems * sizeof(float), stream);

  head_pool_gemm_norm_kernel<<<kM / 64, 256, 0, stream>>>(maps, Wh, bh, out, qn);
  affinity_kernel<<<dim3(4, 4, kB * (kT - 1)), 256, 0, stream>>>(qn, AA);
  softmax_kernel<<<kB * kSteps * kN, 256, 0, stream>>>(AA);
  eye_kernel<<<(int)(atElems / 256), 256, 0, stream>>>(Ata);

  float* cur = Ata; float* nxt = Atb;
  for (int s = 0; s < kSteps; ++s) {
    chain_step_kernel<<<dim3(4, 4, kB), 256, 0, stream>>>(AA, s, cur, nxt);
    float* tmp = cur; cur = nxt; nxt = tmp;
  }

  loss_row_kernel<<<kBN, 256, 0, stream>>>(cur, pl, pa);
  loss_reduce_kernel<<<1, 256, 0, stream>>>(pl, pa, out + kQElems);
}